// DWT3DTiny_88484916232342
// MI455X (gfx1250) — compile-verified
//
#include <hip/hip_runtime.h>

// DWT3DTiny LLL band: out[s,g,v,c] = 2^(-3/2) * sum over 2x2x2 block of
// x[2s+dt, 2g+dh, 2v+dw, c].  x: (32,512,512,8) f32, out: (16,256,256,8) f32.
//
// Roofline: 256 MB read (each input used once) + 32 MB written = 288 MB at
// 23.3 TB/s -> ~12.4 us floor; 67 MFLOP -> 0.23 FLOP/B, purely HBM-bound.
// Load path: 8x global_load_b128 th:TH_LOAD_NT per thread (s_clause'd by the
// compiler), v_pk_add_f32 math. Store path: stage the wave's 512 B in LDS and
// hand writeback to the gfx1250 async DMA engine
// (global_store_async_from_lds_b128, tracked by ASYNCcnt) — zero extra HBM
// traffic, trivial LDS traffic (4 KB/block), wave-private so no barriers.

typedef float v4f __attribute__((ext_vector_type(4)));
// Builtin's pointer params are int32x4 (GCC vector_size style), per hipcc's
// diagnostic: "__attribute__((__vector_size__(4 * sizeof(int)))) int *".
typedef int v4i_vs __attribute__((vector_size(16)));
typedef __attribute__((address_space(1))) v4i_vs* global_v4i_ptr;
typedef __attribute__((address_space(3))) v4i_vs* lds_v4i_ptr;

#define T_IN   32
#define H_IN   512
#define W_IN   512
#define C_IN   8
#define HO     256
#define WO     256
#define ROW4   (W_IN * C_IN / 4)   // float4 per (t,h) row = 1024

#if defined(__gfx1250__) && __has_builtin(__builtin_amdgcn_global_store_async_from_lds_b128)
#define USE_ASYNC_STORE 1
#else
#define USE_ASYNC_STORE 0
#endif

__device__ __forceinline__ void wait_dscnt0() {
#if __has_builtin(__builtin_amdgcn_s_wait_dscnt)
    __builtin_amdgcn_s_wait_dscnt(0);
#else
    asm volatile("s_wait_dscnt 0x0" ::: "memory");
#endif
}

__device__ __forceinline__ void wait_asynccnt0() {
#if __has_builtin(__builtin_amdgcn_s_wait_asynccnt)
    __builtin_amdgcn_s_wait_asynccnt(0);
#else
    asm volatile("s_wait_asynccnt 0x0" ::: "memory");
#endif
}

__global__ __launch_bounds__(256) void dwt3d_haar_lll_kernel(
    const v4f* __restrict__ x, v4f* __restrict__ out)
{
    // Flat output float4 index: n = ((s*HO + g)*WO + v)*2 + q,
    // q selects channels 0..3 (q=0) or 4..7 (q=1).
    const int n = blockIdx.x * 256 + threadIdx.x;

    const int q = n & 1;
    const int v = (n >> 1) & (WO - 1);
    const int g = (n >> 9) & (HO - 1);
    const int s = n >> 17;

    // Input float4 index for (t, h, w=2v|2v+1, 4ch-block q):
    //   (t*H_IN + h)*ROW4 + 4*v + q   (+2 for w = 2v+1)
    const int r0 = ((2 * s) * H_IN + (2 * g)) * ROW4 + 4 * v + q; // t=2s, h=2g
    const int r1 = r0 + ROW4;                                     // h=2g+1
    const int r2 = r0 + H_IN * ROW4;                              // t=2s+1
    const int r3 = r2 + ROW4;

    // 8x non-temporal 128-bit loads (single-use stream: don't pollute L2).
    v4f a0 = __builtin_nontemporal_load(x + r0);
    v4f a1 = __builtin_nontemporal_load(x + r0 + 2);
    v4f b0 = __builtin_nontemporal_load(x + r1);
    v4f b1 = __builtin_nontemporal_load(x + r1 + 2);
    v4f c0 = __builtin_nontemporal_load(x + r2);
    v4f c1 = __builtin_nontemporal_load(x + r2 + 2);
    v4f d0 = __builtin_nontemporal_load(x + r3);
    v4f d1 = __builtin_nontemporal_load(x + r3 + 2);

    v4f acc = (a0 + a1) + (b0 + b1) + ((c0 + c1) + (d0 + d1));
    acc = acc * 0.35355339059327378f;   // 2^(-3/2) = (sqrt(2)/2)^3

#if USE_ASYNC_STORE
    // Stage result in LDS (coalesced ds_store_b128, bank-conflict-free),
    // then let the async DMA engine write it back (ASYNCcnt path).
    __shared__ v4f stage[256];
    stage[threadIdx.x] = acc;
    wait_dscnt0();   // DScnt and ASYNCcnt are unordered: commit LDS first
    __builtin_amdgcn_global_store_async_from_lds_b128(
        (global_v4i_ptr)(void*)(out + n),
        (lds_v4i_ptr)(void*)&stage[threadIdx.x],
        /*offset=*/0, /*cpol=*/0);
    wait_asynccnt0(); // don't release LDS to the next block mid-DMA
#else
    __builtin_nontemporal_store(acc, out + n);
#endif
}

extern "C" void kernel_launch(void* const* d_in, const int* in_sizes, int n_in,
                              void* d_out, int out_size, void* d_ws, size_t ws_size,
                              hipStream_t stream)
{
    (void)in_sizes; (void)n_in; (void)out_size; (void)d_ws; (void)ws_size;

    const v4f* x   = (const v4f*)d_in[0];
    v4f*       out = (v4f*)d_out;

    // Output float4 count: 16*256*256*8/4 = 2,097,152 = 8192 blocks * 256.
    const int total4 = (T_IN / 2) * HO * WO * (C_IN / 4);
    const int block  = 256;                 // 8 wave32 waves
    const int grid   = total4 / block;      // exact: 8192

    dwt3d_haar_lll_kernel<<<grid, block, 0, stream>>>(x, out);
}